// YoloLayer_12275016532328
// MI455X (gfx1250) — compile-verified
//
#include <hip/hip_runtime.h>

// YOLOv4 head for MI455X (gfx1250, wave32).
// in  [B=32, A=3, attrs=85, gy=76, gx=76]  ->  out [B, gy, gx, A, attrs]
// Bandwidth-bound permute+activation. Each block stages one (b,a,gy) slice
// (85 rows x 76 f32, row stride 5776 f32) into LDS with ONE Tensor Data
// Mover descriptor (CDNA5 TDM, TENSORcnt), then emits transposed, coalesced,
// non-temporal stores with the activations applied.

#define GDIM   76
#define ADIM   3
#define ATTRS  85
#define GG     (GDIM * GDIM)       // 5776
#define STRIDE 8.0f                // 608 / 76

typedef __attribute__((ext_vector_type(4))) unsigned int v4u;
typedef __attribute__((ext_vector_type(8))) unsigned int v8u;

__global__ __launch_bounds__(256) void yolo_head_kernel(
    const float* __restrict__ x, float* __restrict__ out) {
  // 85 rows x 76 cols; row = 304 B (16B aligned). Transposed (column) reads
  // have only 2-way bank conflicts (stride 76 words, gcd(76,64)=4, period 16).
  __shared__ __align__(16) float tile[ATTRS * GDIM];

  const int bid = (int)blockIdx.x;
  const int gy  = bid % GDIM;
  const int a   = (bid / GDIM) % ADIM;
  const int b   = bid / (GDIM * ADIM);
  const int tid = (int)threadIdx.x;

  const float* src = x + ((size_t)(b * ADIM + a) * ATTRS) * GG + (size_t)gy * GDIM;

  // ---- Stage tile via Tensor Data Mover: one 2D descriptor per block ----
  if (tid < 32) {                       // wave 0 only: TDM is wave-level
    const unsigned long long ga = (unsigned long long)(size_t)src;
    const unsigned lds_base = (unsigned)(size_t)&tile[0];

    v4u g0;
    g0.x = 1u;                                   // count=1, user desc, no gather
    g0.y = lds_base;                             // lds_addr (bytes)
    g0.z = (unsigned)ga;                         // global_addr[31:0]
    g0.w = (unsigned)(ga >> 32) | (2u << 30);    // global_addr[56:32] | type=2

    v8u g1;
    g1.s0 = 2u << 16;                            // data_size=4B, mask=0, no pad
    g1.s1 = ((unsigned)GG & 0xFFFFu) << 16;      // [63:48] tensor_dim0 lo16
    g1.s2 = ((unsigned)ATTRS << 16) |            // [95:80] tensor_dim1 lo16
            (((unsigned)GG >> 16) & 0xFFFFu);    // [79:64] tensor_dim0 hi16
    g1.s3 = ((unsigned)GDIM << 16);              // [127:112] tile_dim0 = 76
    g1.s4 = (unsigned)ATTRS;                     // [143:128] tile_dim1 = 85
    g1.s5 = (unsigned)GG;                        // tensor_dim0_stride lo32
    g1.s6 = 0u;                                  // stride hi16 | dim1_stride lo
    g1.s7 = 0u;

    v4u g2;                                      // benign 3D/4D identity
    g2.x = 1u;                                   // tensor_dim2 = 1
    g2.y = 1u;                                   // tensor_dim3 = 1
    g2.z = 0u;                                   // tensor_dim2_stride lo
    g2.w = 0u;                                   // tile_dim3 = 0 (unused)

    v4u g3;
    g3.x = 0u;                                   // tensor_dim3_stride lo
    g3.y = 1u << 16;                             // tensor_dim4 = 1
    g3.z = 0u;                                   // tile_dim4 = 0 (unused)
    g3.w = 0u;

    asm volatile("tensor_load_to_lds %0, %1, %2, %3"
                 :: "s"(g0), "s"(g1), "s"(g2), "s"(g3) : "memory");
    __builtin_amdgcn_s_wait_tensorcnt(0);
  }
  __syncthreads();

  // ---- Transform + transposed coalesced NT store ----
  // anchors (grid units) = ALL_ANCHORS[0..2] / 8
  const float aw = (a == 0) ? 1.5f : (a == 1) ? 2.375f : 5.0f;
  const float ah = (a == 0) ? 2.0f : (a == 1) ? 4.5f   : 3.5f;
  const float fgy = (float)gy;

  float* dst = out + ((size_t)(b * GDIM + gy) * GDIM) * (ADIM * ATTRS) + a * ATTRS;

  // e = gx*85 + c, stride 256 = 3*85 + 1  ->  incremental (gx, c) update,
  // no per-iteration division (cuts the v_mul_hi/s_mul magic-div chains).
  int gx = tid / ATTRS;
  int c  = tid - gx * ATTRS;
  for (int e = tid; e < GDIM * ATTRS; e += 256) {
    const float v = tile[c * GDIM + gx];
    float r;
    if (c >= 4) {
      r = __builtin_amdgcn_rcpf(1.0f + __expf(-v));            // sigmoid
    } else if (c == 0) {
      r = (__builtin_amdgcn_rcpf(1.0f + __expf(-v)) + (float)gx) * STRIDE;
    } else if (c == 1) {
      r = (__builtin_amdgcn_rcpf(1.0f + __expf(-v)) + fgy) * STRIDE;
    } else {
      const float an = (c == 2) ? aw : ah;
      r = (__expf(v) * an) * STRIDE;
    }
    __builtin_nontemporal_store(r, &dst[(size_t)gx * (ADIM * ATTRS) + c]);

    c += 1; gx += 3;
    if (c >= ATTRS) { c -= ATTRS; gx += 1; }
  }
}

extern "C" void kernel_launch(void* const* d_in, const int* in_sizes, int n_in,
                              void* d_out, int out_size, void* d_ws, size_t ws_size,
                              hipStream_t stream) {
  const float* x = (const float*)d_in[0];
  (void)in_sizes; (void)n_in; (void)d_ws; (void)ws_size; (void)out_size;
  float* out = (float*)d_out;

  const int blocks = 32 * ADIM * GDIM;   // one block per (b, a, gy) = 7296
  yolo_head_kernel<<<dim3(blocks), dim3(256), 0, stream>>>(x, out);
}